// CTClass_644245094854
// MI455X (gfx1250) — compile-verified
//
#include <hip/hip_runtime.h>
#include <math.h>

#ifndef M_PI
#define M_PI 3.14159265358979323846
#endif

#define SIG   512
#define NA    45
#define DIAG  725           // ceil(sqrt(2)*512)
#define PADB  106           // PAD_BEFORE
#define BATCH 2
#define YCHUNKS 5           // 725 = 5 * 145
#define YCHUNK_LEN 145

typedef __attribute__((ext_vector_type(2))) float v2f;
typedef __attribute__((ext_vector_type(8))) float v8f;

__device__ __forceinline__ int clampi(int v, int lo, int hi) {
    return min(max(v, lo), hi);
}

// Branchless sample of the implicitly-padded image. The padded pixel (yi, xi)
// is nonzero only inside [PADB, PADB+512)^2. We clamp the index so the load is
// always in-bounds (issues unconditionally, no exec-mask branch) and zero the
// result via select — identical semantics to grid_sample clip+valid*weight.
__device__ __forceinline__ float tap_img(const float* __restrict__ imgb, int xi, int yi) {
    int ox = xi - PADB;
    int oy = yi - PADB;
    bool valid = ((unsigned)ox < (unsigned)SIG) & ((unsigned)oy < (unsigned)SIG);
    int xc = clampi(ox, 0, SIG - 1);
    int yc = clampi(oy, 0, SIG - 1);
    float v = imgb[yc * SIG + xc];
    return valid ? v : 0.0f;
}

// ---------------------------------------------------------------------------
// Kernel 1: fused rotate + row-sum (forward Radon), partial over y-chunks.
// Thread = (chunk, b, a, x). partial[chunk][(b*NA+a)*DIAG + x] = sum over its
// 145 rotated rows of the bilinear sample at column x.
// ---------------------------------------------------------------------------
__global__ __launch_bounds__(256)
void radon_partial_kernel(const float* __restrict__ img, float* __restrict__ partial)
{
    const int total = BATCH * NA * DIAG;
    int idx = blockIdx.x * blockDim.x + threadIdx.x;
    if (idx >= total * YCHUNKS) return;

    int c   = idx / total;
    int r   = idx - c * total;
    int x   = r % DIAG;
    int a   = (r / DIAG) % NA;
    int b   = r / (DIAG * NA);

    float th = (float)((double)a * M_PI / 45.0);   // theta = a * 4 degrees
    float ct = cosf(th), st = sinf(th);

    const float step = 2.0f / (float)(DIAG - 1);
    const float half = 0.5f * (float)(DIAG - 1);
    float cx = fmaf((float)x, step, -1.0f);

    // px = (ct*cx + st*cy + 1)*half = Ax + (half*st)*cy
    // py = (ct*cy - st*cx + 1)*half = Ay + (half*ct)*cy
    float Ax = fmaf(half * ct, cx, half);
    float Ay = fmaf(-half * st, cx, half);
    float Sx = half * st;
    float Sy = half * ct;

    const float* imgb = img + (size_t)b * SIG * SIG;

    float acc = 0.0f;
    int ybeg = c * YCHUNK_LEN;
    for (int y = ybeg; y < ybeg + YCHUNK_LEN; ++y) {
        float cy = fmaf((float)y, step, -1.0f);
        float px = fmaf(Sx, cy, Ax);
        float py = fmaf(Sy, cy, Ay);
        float fx = floorf(px), fy = floorf(py);
        float wx = px - fx,    wy = py - fy;
        int x0 = (int)fx, y0 = (int)fy;

        float v00 = tap_img(imgb, x0,     y0);
        float v10 = tap_img(imgb, x0 + 1, y0);
        float v01 = tap_img(imgb, x0,     y0 + 1);
        float v11 = tap_img(imgb, x0 + 1, y0 + 1);

        float top = v00 + (v10 - v00) * wx;
        float bot = v01 + (v11 - v01) * wx;
        acc += top + (bot - top) * wy;
    }
    partial[(size_t)c * total + r] = acc;
}

// ---------------------------------------------------------------------------
// Kernel 2: deterministic fixed-order reduce over y-chunks, then subtract the
// measured sinogram y[b,0,t,a]. Output layout sino[b][a][t] (angle-major) so
// each angle's row is contiguous for the backprojection taps.
// ---------------------------------------------------------------------------
__global__ __launch_bounds__(256)
void radon_reduce_kernel(const float* __restrict__ partial,
                         const float* __restrict__ ysino,
                         float* __restrict__ sino)
{
    const int total = BATCH * NA * DIAG;
    int idx = blockIdx.x * blockDim.x + threadIdx.x;
    if (idx >= total) return;

    int t = idx % DIAG;
    int b = idx / (DIAG * NA);

    float s = 0.0f;
    #pragma unroll
    for (int c = 0; c < YCHUNKS; ++c)
        s += partial[(size_t)c * total + idx];

    // y shape (B,1,DIAG,NA): flat b*DIAG*NA + t*NA + a
    int a = (idx / DIAG) % NA;
    s -= ysino[(size_t)b * DIAG * NA + (size_t)t * NA + a];
    sino[idx] = s;   // idx == (b*NA + a)*DIAG + t
}

// ---------------------------------------------------------------------------
// Kernel 3: backprojection with WMMA f32 16x16x4 angle-reduction.
// Wave owns 16 consecutive output pixels (B/C/D columns = lanes mod 16); each
// chunk of 4 angles forms the B-matrix k-rows, 2 per lane-half. A = all-ones,
// so D accumulates per-pixel angle sums (invariant to the k<->slot mapping).
// All taps are branchless: indices clamped, validity folded into the lerp
// weights, so EXEC stays all-ones around every WMMA.
// ---------------------------------------------------------------------------
__global__ __launch_bounds__(256)
void backproject_wmma_kernel(const float* __restrict__ sino, float* __restrict__ out)
{
    __shared__ float s_ct[48];
    __shared__ float s_st[48];
    if (threadIdx.x < 48) {
        int a = threadIdx.x;
        float th = (a < NA) ? (float)((double)a * M_PI / 45.0) : 0.0f;
        s_ct[a] = cosf(th);
        s_st[a] = sinf(th);
    }
    __syncthreads();

    int wave = blockIdx.x * (blockDim.x >> 5) + (threadIdx.x >> 5);
    int lane = threadIdx.x & 31;
    int n = lane & 15;        // B/C/D column owned by this lane
    int g = lane >> 4;        // lane half -> which pair of k-rows we supply

    int xtile = wave & 31;            // 32 tiles of 16 px -> 512 columns
    int yy    = (wave >> 5) & 511;    // output row
    int b     = wave >> 14;           // batch

    int xx = xtile * 16 + n;
    const float step = 2.0f / (float)(DIAG - 1);
    const float half = 0.5f * (float)(DIAG - 1);
    float cx = fmaf((float)(xx + PADB), step, -1.0f);
    float cy = fmaf((float)(yy + PADB), step, -1.0f);

    const float* sb = sino + (size_t)b * NA * DIAG;

    v8f acc = {};
    v2f ones; ones.x = 1.0f; ones.y = 1.0f;

    for (int cc = 0; cc < 12; ++cc) {
        int abase = cc * 4 + g * 2;
        v2f bm;
        #pragma unroll
        for (int k = 0; k < 2; ++k) {
            int a = abase + k;
            float avalid = (a < NA) ? 1.0f : 0.0f;  // tail angles contribute 0
            int ac = min(a, NA - 1);                // keep loads in-bounds

            float t  = s_ct[ac] * cx - s_st[ac] * cy;
            float pt = (t + 1.0f) * half;
            float ft = floorf(pt);
            float wt = pt - ft;
            int ti = (int)ft;

            float w0 = (1.0f - wt) * ((((unsigned)ti)       < (unsigned)DIAG) ? avalid : 0.0f);
            float w1 = wt          * ((((unsigned)(ti + 1)) < (unsigned)DIAG) ? avalid : 0.0f);
            int t0c = clampi(ti,     0, DIAG - 1);
            int t1c = clampi(ti + 1, 0, DIAG - 1);
            const float* row = sb + ac * DIAG;
            float contrib = row[t0c] * w0 + row[t1c] * w1;

            if (k == 0) bm.x = contrib; else bm.y = contrib;
        }
        // D = ones(16x4) x B(4x16) + C : every D row = per-column angle sum
        acc = __builtin_amdgcn_wmma_f32_16x16x4_f32(
            /*neg_a=*/false, ones, /*neg_b=*/false, bm,
            /*c_mod=*/(short)0, acc, /*reuse_a=*/false, /*reuse_b=*/false);
    }

    if (g == 0) {  // lanes 0..15 carry columns 0..15; row M=0 lives in acc[0]
        const float scale = (float)(M_PI / (2.0 * (double)NA));
        out[(size_t)b * SIG * SIG + (size_t)yy * SIG + xx] = acc[0] * scale;
    }
}

// ---------------------------------------------------------------------------
extern "C" void kernel_launch(void* const* d_in, const int* in_sizes, int n_in,
                              void* d_out, int out_size, void* d_ws, size_t ws_size,
                              hipStream_t stream)
{
    const float* x = (const float*)d_in[0];   // (2,1,512,512) f32
    const float* y = (const float*)d_in[1];   // (2,1,725,45)  f32
    float* out = (float*)d_out;               // (2,1,512,512) f32

    const int total = BATCH * NA * DIAG;                // 65,250
    float* partial = (float*)d_ws;                      // YCHUNKS * total floats
    float* sino    = partial + (size_t)YCHUNKS * total; // total floats

    int n1 = total * YCHUNKS;                           // 326,250 threads
    radon_partial_kernel<<<(n1 + 255) / 256, 256, 0, stream>>>(x, partial);
    radon_reduce_kernel<<<(total + 255) / 256, 256, 0, stream>>>(partial, y, sino);

    // 2 batches * 512 rows * 32 tiles = 32768 waves; 8 waves/block -> 4096 blocks
    backproject_wmma_kernel<<<4096, 256, 0, stream>>>(sino, out);
}